// RecPolicy_6811818131651
// MI455X (gfx1250) — compile-verified
//
#include <hip/hip_runtime.h>
#include <hip/hip_bf16.h>

// CDNA5 / gfx1250, wave32. Recurrent policy net, H=4, B=1M.
// Recurrence kept transposed: h.T is a WMMA B-fragment, 4xK weights are
// constant A-fragments (rows>=4 zero), biases folded into C. Every matmul
// stage runs on V_WMMA_F32_16X16X4_F32 (exact fp32); tanh runs on the
// transcendental VALU pipe and co-executes with the XDL WMMA pipe.
// One wave = 16 batch rows, 30 WMMAs, fully branch-free inner math.

typedef float v2f __attribute__((ext_vector_type(2)));
typedef float v4f __attribute__((ext_vector_type(4)));
typedef float v8f __attribute__((ext_vector_type(8)));

#define B_TOTAL        1048576
#define ROWS_PER_WAVE  16
#define WAVES_PER_BLK  8
#define ROWS_PER_BLK   (ROWS_PER_WAVE * WAVES_PER_BLK)

static_assert(B_TOTAL % ROWS_PER_BLK == 0, "tail not handled");

__device__ __forceinline__ float fast_tanh(float x) {
#if __has_builtin(__builtin_amdgcn_tanhf)
    return __builtin_amdgcn_tanhf(x);   // v_tanh_f32
#else
    return tanhf(x);
#endif
}

// Swap lane halves (lane ^= 16) with an immediate-mode swizzle: group-of-32,
// and_mask=0x1f, or_mask=0, xor_mask=0x10  ->  offset 0x401F (SWAPX16).
__device__ __forceinline__ float swap16(float x) {
    return __int_as_float(
        __builtin_amdgcn_ds_swizzle(__float_as_int(x), 0x401F));
}

// D = A(16x4 f32) * B(4x16 f32) + C(16x16 f32)
__device__ __forceinline__ v8f wmma4(v2f a, v2f b, v8f c) {
    return __builtin_amdgcn_wmma_f32_16x16x4_f32(false, a, false, b,
                                                 (short)0, c, false, false);
}

// A-fragment (16x4, rows>=4 zero) from row-major (4 x ld) weight, columns
// [koff, koff+kin). Branch-free: clamp indices in-bounds, mask with 0/1.
__device__ __forceinline__ v2f load_a(const float* W, int ld, int koff,
                                      int kin, int m, int half) {
    const int   mr   = (m < 4) ? m : 3;
    const float keep = (m < 4) ? 1.f : 0.f;
    const int   k0   = half * 2;
    const float m0   = (k0     < kin) ? keep : 0.f;
    const float m1   = (k0 + 1 < kin) ? keep : 0.f;
    const int   i0   = mr * ld + koff + ((k0     < kin) ? k0     : 0);
    const int   i1   = mr * ld + koff + ((k0 + 1 < kin) ? k0 + 1 : 0);
    v2f a;
    a.x = W[i0] * m0;
    a.y = W[i1] * m1;
    return a;
}

// C-fragment: bias in rows 0..3 (VGPRs 0..3, lanes 0..15), zero elsewhere.
// Uniform loads (scalarizable) + mask multiply, no exec branching.
__device__ __forceinline__ v8f load_bias_c(const float* ba, const float* bb,
                                           int half) {
    const float keep = (half == 0) ? 1.f : 0.f;
    v8f c = {0.f, 0.f, 0.f, 0.f, 0.f, 0.f, 0.f, 0.f};
#pragma unroll
    for (int r = 0; r < 4; ++r)
        c[r] = (ba[r] + (bb ? bb[r] : 0.f)) * keep;
    return c;
}

__device__ __forceinline__ void tanh4(v8f& d) {
    d[0] = fast_tanh(d[0]); d[1] = fast_tanh(d[1]);
    d[2] = fast_tanh(d[2]); d[3] = fast_tanh(d[3]);
}

// D rows 0..3 (h_new.T) -> B-fragment: K0/K1 from D0/D1 (lanes<16),
// K2/K3 mirrored from D2/D3 into lanes>=16 via half-swap swizzle.
__device__ __forceinline__ v2f d_to_b(v8f d, int half) {
    const float d2s = swap16(d[2]);
    const float d3s = swap16(d[3]);
    v2f h;
    h.x = half ? d2s : d[0];
    h.y = half ? d3s : d[1];
    return h;
}

__global__ __launch_bounds__(256) void rec_policy_kernel(
    const float* __restrict__ x,
    const float* __restrict__ Wih_up, const float* __restrict__ Whh_up,
    const float* __restrict__ bih_up, const float* __restrict__ bhh_up,
    const float* __restrict__ W1,     const float* __restrict__ b1,
    const float* __restrict__ W2,     const float* __restrict__ b2,
    const float* __restrict__ Wih_dn, const float* __restrict__ Whh_dn,
    const float* __restrict__ bih_dn, const float* __restrict__ bhh_dn,
    const float* __restrict__ Wo,     const float* __restrict__ bo,
    float* __restrict__ out)
{
    __shared__ float lds_x[WAVES_PER_BLK][ROWS_PER_WAVE * 18];                 // 9216 B
    __shared__ __align__(16) float lds_o[WAVES_PER_BLK][ROWS_PER_WAVE * 7];    // 3584 B

    const int lane = threadIdx.x & 31;
    const int wid  = threadIdx.x >> 5;
    const int half = lane >> 4;       // which 16-lane half
    const int n    = lane & 15;       // batch column within tile / weight row
    const int tile = blockIdx.x * WAVES_PER_BLK + wid;

    // ---- coalesced input stage: 16 rows x 18 f32 = 288 = 9 x 32 dwords ----
    {
        const float* src = x + (size_t)tile * (ROWS_PER_WAVE * 18);
#pragma unroll
        for (int i = 0; i < 9; ++i)
            lds_x[wid][i * 32 + lane] = __builtin_nontemporal_load(&src[i * 32 + lane]);
    }

    // ---- constant weight A-fragments & bias C-fragments ----
    const v2f A_ih_up = load_a(Wih_up, 2, 0, 2, n, half);
    const v2f A_hh_up = load_a(Whh_up, 4, 0, 4, n, half);
    const v2f A_W1a   = load_a(W1,     8, 0, 4, n, half);  // obs part
    const v2f A_W1b   = load_a(W1,     8, 4, 4, n, half);  // h_last part
    const v2f A_W2    = load_a(W2,     4, 0, 4, n, half);
    const v2f A_ih_dn = load_a(Wih_dn, 4, 0, 4, n, half);
    const v2f A_hh_dn = load_a(Whh_dn, 4, 0, 4, n, half);

    const v8f C_up = load_bias_c(bih_up, bhh_up, half);
    const v8f C_b1 = load_bias_c(b1, nullptr, half);
    const v8f C_b2 = load_bias_c(b2, nullptr, half);
    const v8f C_dn = load_bias_c(bih_dn, bhh_dn, half);

    const float wo_lo = half ? Wo[2] : Wo[0];   // k = 0/2 per half
    const float wo_hi = half ? Wo[3] : Wo[1];   // k = 1/3 per half
    const float wo_j  = Wo[4];
    const float bo_s  = bo[0];

    // ---- pull this lane's full 18-float row into registers (9 x ds_load_b64,
    //      unconditional -> no exec-masked loads anywhere below) ----
    float rr[18];
    {
        const float2* row2 = (const float2*)&lds_x[wid][n * 18];  // 8B aligned
#pragma unroll
        for (int i = 0; i < 9; ++i) {
            const float2 q = row2[i];
            rr[2 * i]     = q.x;
            rr[2 * i + 1] = q.y;
        }
    }

    // ---- up pass: t = 6 .. 0 (seq reversed), h = tanh(Wih*inp + Whh*h + b) ----
    v2f h;
    v2f hup[7];
    {   // t = 6 peeled: h starts at zero, skip the Whh WMMA entirely
        v2f bi;
        bi.x = half ? 0.f : rr[4 + 6];
        bi.y = half ? 0.f : rr[11 + 6];
        v8f d = wmma4(A_ih_up, bi, C_up);
        tanh4(d);
        h = d_to_b(d, half);
        hup[6] = h;
    }
#pragma unroll
    for (int t = 5; t >= 0; --t) {
        v2f bi;                                  // inp.T as B: K0=j_t, K1=jd_t
        bi.x = half ? 0.f : rr[4 + t];
        bi.y = half ? 0.f : rr[11 + t];
        v8f d = wmma4(A_ih_up, bi, C_up);
        d = wmma4(A_hh_up, h, d);
        tanh4(d);
        h = d_to_b(d, half);
        hup[t] = h;                              // h_up[t] (forward index)
    }

    // ---- bottleneck MLP: tanh(W2 * tanh(W1 * [obs; h_last] + b1) + b2) ----
    {
        v2f bobs;                                // obs.T as B: K = 0..3
        bobs.x = half ? rr[2] : rr[0];
        bobs.y = half ? rr[3] : rr[1];
        v8f d = wmma4(A_W1a, bobs, C_b1);
        d = wmma4(A_W1b, h, d);
        tanh4(d);
        h = d_to_b(d, half);
        d = wmma4(A_W2, h, C_b2);
        tanh4(d);
        h = d_to_b(d, half);
    }

    // ---- down pass: act_t from current h, then h update ----
    float acts[7];
#pragma unroll
    for (int t = 0; t < 7; ++t) {
        // act = Wo[0..3].h + Wo[4]*j_t + bo : per-half partial, half-swap sum
        float p = wo_lo * h.x + wo_hi * h.y;
        p += swap16(p);
        acts[t] = p + wo_j * rr[4 + t] + bo_s;   // valid on half==0 lanes

        v8f d = wmma4(A_ih_dn, hup[t], C_dn);
        d = wmma4(A_hh_dn, h, d);
        tanh4(d);
        h = d_to_b(d, half);
    }

    // ---- output: stage 16 rows x 7 acts, then 28 coalesced float4 stores ----
    if (half == 0) {
#pragma unroll
        for (int t = 0; t < 7; ++t)
            lds_o[wid][n * 7 + t] = acts[t];
    }
    if (lane < 28) {
        const v4f v = *(const v4f*)&lds_o[wid][lane * 4];
        v4f* dst = (v4f*)(out + (size_t)tile * (ROWS_PER_WAVE * 7) + lane * 4);
        __builtin_nontemporal_store(v, dst);
    }
}

extern "C" void kernel_launch(void* const* d_in, const int* in_sizes, int n_in,
                              void* d_out, int out_size, void* d_ws, size_t ws_size,
                              hipStream_t stream) {
    (void)in_sizes; (void)n_in; (void)out_size; (void)d_ws; (void)ws_size;
    const float* x      = (const float*)d_in[0];
    const float* Wih_up = (const float*)d_in[1];
    const float* Whh_up = (const float*)d_in[2];
    const float* bih_up = (const float*)d_in[3];
    const float* bhh_up = (const float*)d_in[4];
    const float* W1     = (const float*)d_in[5];
    const float* b1     = (const float*)d_in[6];
    const float* W2     = (const float*)d_in[7];
    const float* b2     = (const float*)d_in[8];
    const float* Wih_dn = (const float*)d_in[9];
    const float* Whh_dn = (const float*)d_in[10];
    const float* bih_dn = (const float*)d_in[11];
    const float* bhh_dn = (const float*)d_in[12];
    const float* Wo     = (const float*)d_in[13];
    const float* bo     = (const float*)d_in[14];
    float* out = (float*)d_out;

    dim3 grid(B_TOTAL / ROWS_PER_BLK);   // 8192 blocks x 256 threads (8 waves)
    rec_policy_kernel<<<grid, dim3(256), 0, stream>>>(
        x, Wih_up, Whh_up, bih_up, bhh_up, W1, b1, W2, b2,
        Wih_dn, Whh_dn, bih_dn, bhh_dn, Wo, bo, out);
}